// GaeNet_51453708206756
// MI455X (gfx1250) — compile-verified
//
#include <hip/hip_runtime.h>
#include <hip/hip_bf16.h>

typedef float v2f __attribute__((ext_vector_type(2)));
typedef float v8f __attribute__((ext_vector_type(8)));

#define XDIM 972
#define XOFF 588
#define K1   384
#define F1   128
#define F2   64

// ---------------- degree / normalization ----------------

__global__ void k_init_deg(float* __restrict__ deg, int n) {
    int i = blockIdx.x * blockDim.x + threadIdx.x;
    if (i < n) deg[i] = 1.0f;                 // self-loop contributes 1
}

__global__ void k_deg_accum(const long long* __restrict__ col, float* __restrict__ deg, int E) {
    int e = blockIdx.x * blockDim.x + threadIdx.x;
    if (e < E) atomicAdd(&deg[(int)col[e]], 1.0f);
}

__global__ void k_dinv(float* __restrict__ deg, int n) {
    int i = blockIdx.x * blockDim.x + threadIdx.x;
    if (i < n) {
        float d = deg[i];
        deg[i] = d > 0.0f ? rsqrtf(d) : 0.0f; // in-place: deg becomes dinv
    }
}

// ---------------- GEMM1: h = x[:,588:972] @ W1  (M=100000, K=384, N=128) ----------------
// One wave per 16x16 C tile. 8 waves/block cover the 8 N-tiles of one M-tile.

__global__ void k_gemm1(const float* __restrict__ x, const float* __restrict__ W1,
                        float* __restrict__ h) {
    const int lane  = threadIdx.x & 31;
    const int wave  = threadIdx.x >> 5;      // 0..7 -> N tile
    const int l16   = lane & 15;
    const int khalf = lane >> 4;             // 0: K pair {0,1}; 1: K pair {2,3}
    const int m     = blockIdx.x * 16 + l16; // A-matrix row for this lane
    const int ncol  = wave * 16 + l16;       // B/C column for this lane

    const float* __restrict__ arow = x + (size_t)m * XDIM + XOFF;

    v8f acc = {};
    for (int k = 0; k < K1; k += 4) {
        const int kc = k + 2 * khalf;
        v2f a = *(const v2f*)(arow + kc);    // A[m][kc], A[m][kc+1]
        v2f b;
        b.x = W1[(size_t)(kc    ) * F1 + ncol];
        b.y = W1[(size_t)(kc + 1) * F1 + ncol];
        acc = __builtin_amdgcn_wmma_f32_16x16x4_f32(false, a, false, b,
                                                    (short)0, acc, false, false);
    }

    // C/D layout: VGPR r -> M = r + 8*khalf, N = lane&15
    const int mbase = blockIdx.x * 16 + 8 * khalf;
#pragma unroll
    for (int r = 0; r < 8; ++r)
        h[(size_t)(mbase + r) * F1 + ncol] = acc[r];
}

// ---------------- GEMM2: h2 = relu(agg1 + b1) @ W2  (M=100000, K=128, N=64) ----------------

__global__ void k_gemm2(const float* __restrict__ agg1, const float* __restrict__ b1,
                        const float* __restrict__ W2, float* __restrict__ h2) {
    const int lane  = threadIdx.x & 31;
    const int wave  = threadIdx.x >> 5;      // 0..3 -> N tile
    const int l16   = lane & 15;
    const int khalf = lane >> 4;
    const int m     = blockIdx.x * 16 + l16;
    const int ncol  = wave * 16 + l16;

    const float* __restrict__ arow = agg1 + (size_t)m * F1;

    v8f acc = {};
    for (int k = 0; k < F1; k += 4) {
        const int kc = k + 2 * khalf;
        v2f a;
        a.x = fmaxf(arow[kc    ] + b1[kc    ], 0.0f);  // fused bias + ReLU
        a.y = fmaxf(arow[kc + 1] + b1[kc + 1], 0.0f);
        v2f b;
        b.x = W2[(size_t)(kc    ) * F2 + ncol];
        b.y = W2[(size_t)(kc + 1) * F2 + ncol];
        acc = __builtin_amdgcn_wmma_f32_16x16x4_f32(false, a, false, b,
                                                    (short)0, acc, false, false);
    }

    const int mbase = blockIdx.x * 16 + 8 * khalf;
#pragma unroll
    for (int r = 0; r < 8; ++r)
        h2[(size_t)(mbase + r) * F2 + ncol] = acc[r];
}

// ---------------- self-loop init: agg[i] = h[i] * dinv[i]^2 (also re-inits agg each call) ----

__global__ void k_selfloop(const float* __restrict__ hsrc, const float* __restrict__ dinv,
                           float* __restrict__ agg, int n, int f4) {
    int t = blockIdx.x * blockDim.x + threadIdx.x;
    int total = n * f4;
    if (t >= total) return;
    int i = t / f4;
    float w = dinv[i] * dinv[i];
    float4 v = ((const float4*)hsrc)[t];
    v.x *= w; v.y *= w; v.z *= w; v.w *= w;
    ((float4*)agg)[t] = v;
}

// ---------------- edge scatter, 128 features: one wave per edge, float4 per lane ----------

__global__ void k_scatter1(const long long* __restrict__ ei, const float* __restrict__ h,
                           const float* __restrict__ dinv, float* __restrict__ agg, int E) {
    int wid  = (int)((blockIdx.x * (size_t)blockDim.x + threadIdx.x) >> 5);
    int lane = threadIdx.x & 31;
    if (wid >= E) return;
    int r = (int)ei[wid];
    int c = (int)ei[(size_t)E + wid];
    float w = dinv[r] * dinv[c];
    float4 v = ((const float4*)(h + (size_t)r * F1))[lane];
    float* dst = agg + (size_t)c * F1 + lane * 4;
    atomicAdd(dst + 0, v.x * w);
    atomicAdd(dst + 1, v.y * w);
    atomicAdd(dst + 2, v.z * w);
    atomicAdd(dst + 3, v.w * w);
}

// ---------------- edge scatter, 64 features: one wave per edge, float2 per lane -----------

__global__ void k_scatter2(const long long* __restrict__ ei, const float* __restrict__ h2,
                           const float* __restrict__ dinv, float* __restrict__ agg, int E) {
    int wid  = (int)((blockIdx.x * (size_t)blockDim.x + threadIdx.x) >> 5);
    int lane = threadIdx.x & 31;
    if (wid >= E) return;
    int r = (int)ei[wid];
    int c = (int)ei[(size_t)E + wid];
    float w = dinv[r] * dinv[c];
    float2 v = ((const float2*)(h2 + (size_t)r * F2))[lane];
    float* dst = agg + (size_t)c * F2 + lane * 2;
    atomicAdd(dst + 0, v.x * w);
    atomicAdd(dst + 1, v.y * w);
}

// ---------------- edge dot: logits[e] = sum_f (z[a][f]+b2[f])*(z[b][f]+b2[f]) -------------

__global__ void k_dot(const long long* __restrict__ eli, const float* __restrict__ z,
                      const float* __restrict__ b2, float* __restrict__ out, int L) {
    int wid  = (int)((blockIdx.x * (size_t)blockDim.x + threadIdx.x) >> 5);
    int lane = threadIdx.x & 31;
    if (wid >= L) return;
    int a = (int)eli[wid];
    int b = (int)eli[(size_t)L + wid];
    float2 za = ((const float2*)(z + (size_t)a * F2))[lane];
    float2 zb = ((const float2*)(z + (size_t)b * F2))[lane];
    float2 bb = ((const float2*)b2)[lane];
    float s = (za.x + bb.x) * (zb.x + bb.x) + (za.y + bb.y) * (zb.y + bb.y);
#pragma unroll
    for (int off = 16; off > 0; off >>= 1)
        s += __shfl_xor(s, off, 32);
    if (lane == 0) out[wid] = s;
}

// ---------------- host launcher ----------------

extern "C" void kernel_launch(void* const* d_in, const int* in_sizes, int n_in,
                              void* d_out, int out_size, void* d_ws, size_t ws_size,
                              hipStream_t stream) {
    (void)n_in; (void)out_size; (void)ws_size;

    const float*      x   = (const float*)d_in[0];
    const long long*  ei  = (const long long*)d_in[1];   // int64 (2,E) row-major
    const long long*  eli = (const long long*)d_in[2];   // int64 (2,L)
    const float*      W1  = (const float*)d_in[3];       // (384,128)
    const float*      b1  = (const float*)d_in[4];       // (128,)
    const float*      W2  = (const float*)d_in[5];       // (128,64)
    const float*      b2  = (const float*)d_in[6];       // (64,)
    float*            out = (float*)d_out;               // (L,)

    const int n = in_sizes[0] / XDIM;   // 100000 (divisible by 16)
    const int E = in_sizes[1] / 2;      // 1600000
    const int L = in_sizes[2] / 2;      // 1000000

    // Workspace layout (floats), 64-element aligned
    float* ws    = (float*)d_ws;
    size_t off   = 0;
    float* dinv  = ws + off; off += (size_t)((n + 63) / 64) * 64;  // n
    float* h     = ws + off; off += (size_t)n * F1;                // 12.8M
    float* agg1  = ws + off; off += (size_t)n * F1;                // 12.8M
    float* h2    = ws + off; off += (size_t)n * F2;                // 6.4M
    float* agg2  = ws + off;                                       // 6.4M

    const int TB = 256;

    // 1) degrees (with self-loop) -> dinv
    k_init_deg <<<(n + TB - 1) / TB, TB, 0, stream>>>(dinv, n);
    k_deg_accum<<<(E + TB - 1) / TB, TB, 0, stream>>>(ei + E, dinv, E);
    k_dinv     <<<(n + TB - 1) / TB, TB, 0, stream>>>(dinv, n);

    // 2) layer 1
    k_gemm1    <<<n / 16, 256, 0, stream>>>(x, W1, h);
    k_selfloop <<<((n * (F1 / 4)) + TB - 1) / TB, TB, 0, stream>>>(h, dinv, agg1, n, F1 / 4);
    k_scatter1 <<<(E + 7) / 8, 256, 0, stream>>>(ei, h, dinv, agg1, E);

    // 3) layer 2 (bias+ReLU fused into GEMM2 A-operand)
    k_gemm2    <<<n / 16, 128, 0, stream>>>(agg1, b1, W2, h2);
    k_selfloop <<<((n * (F2 / 4)) + TB - 1) / TB, TB, 0, stream>>>(h2, dinv, agg2, n, F2 / 4);
    k_scatter2 <<<(E + 7) / 8, 256, 0, stream>>>(ei, h2, dinv, agg2, E);

    // 4) link logits (b2 fused into dot)
    k_dot      <<<(L + 7) / 8, 256, 0, stream>>>(eli, agg2, b2, out, L);
}